// Repro_11879879542592
// MI455X (gfx1250) — compile-verified
//
#include <hip/hip_runtime.h>
#include <hip/hip_bf16.h>

// ---------------------------------------------------------------------------
// Reference: QKV GEMM (V third is dead) + interleaved RoPE + per-head Q@K^T.
//   X: (4096, 768)  W: (2304, 768)  Cos/Sin: (4096, 32)
//   Out: (24, 2048, 2048) fp32  -> 402 MB store dominates (HBM-bound, ~17us)
// fp32 V_WMMA_F32_16X16X4_F32 everywhere (matches reference precision).
// ---------------------------------------------------------------------------

typedef __attribute__((ext_vector_type(2))) float v2f;
typedef __attribute__((ext_vector_type(8))) float v8f;
typedef int v4i_vs __attribute__((vector_size(16)));   // matches builtin param

#define SEQ     2048
#define BATCH   2
#define HID     768
#define NHEAD   12
#define HDIM    64
#define M1      (SEQ * BATCH)   // 4096
#define NPACK   (NHEAD * 128)   // 1536 packed q|k columns

#define GLOBAL_AS __attribute__((address_space(1)))
#define LDS_AS    __attribute__((address_space(3)))

// --- CDNA5 async global->LDS staging (ASYNCcnt path), with sync fallback ---
__device__ __forceinline__ void copy16_to_lds(const float* g, float* l) {
#if __has_builtin(__builtin_amdgcn_global_load_async_to_lds_b128)
    // generic->AS casts via integer (low 32 bits of a generic LDS pointer are
    // the LDS offset; AS1 keeps the full 64-bit VA)
    __builtin_amdgcn_global_load_async_to_lds_b128(
        (GLOBAL_AS v4i_vs*)(uintptr_t)g,
        (LDS_AS v4i_vs*)(uint32_t)(uintptr_t)l,
        0, 0);
#else
    const float4 v = *(const float4*)g;
    l[0] = v.x; l[1] = v.y; l[2] = v.z; l[3] = v.w;
#endif
}

template <int N>
__device__ __forceinline__ void wait_async_le() {
#if __has_builtin(__builtin_amdgcn_global_load_async_to_lds_b128)
#if __has_builtin(__builtin_amdgcn_s_wait_asynccnt)
    __builtin_amdgcn_s_wait_asynccnt((unsigned short)N);
#else
    asm volatile("s_wait_asynccnt %0" :: "i"(N) : "memory");
#endif
#endif
}

// ===========================================================================
// Kernel 1: fused q|k projection + RoPE -> Qw/Kw in (B*NH, S, 64) layout.
// Block 256 thr (8 waves), tile 64(M) x 128(N), K staged 16 wide,
// double-buffered LDS, async-to-LDS staging overlapped with WMMA.
// Wave (wm 0..1, wn 0..3) owns a 32x32 patch = 4 WMMA accumulators.
// A and B both kept [row][k] in LDS (row stride 20 floats: 16B-aligned,
// 20*i mod 64 distinct for i<16 -> bank-conflict-free fragment reads).
// ===========================================================================
__global__ __launch_bounds__(256) void qkv_rope_kernel(
    const float* __restrict__ X,
    const float* __restrict__ W,
    const float* __restrict__ Cos,
    const float* __restrict__ Sin,
    float* __restrict__ Qw,
    float* __restrict__ Kw)
{
    __shared__ float As[2][64][20];    // [buf][m][k]
    __shared__ float Bs[2][128][20];   // [buf][n][k]

    const int tid  = threadIdx.x;
    const int lane = tid & 31;
    const int wave = tid >> 5;

    const int m0 = blockIdx.x * 64;    // 64 blocks over M=4096
    const int n0 = blockIdx.y * 128;   // 12 blocks: one q|k head pair each

    const int wm = wave & 1;           // 32 rows per wave
    const int wn = wave >> 1;          // 32 cols per wave
    const int lrow  = lane & 15;
    const int khalf = (lane >> 4) << 1;

    // --- staging assignment (3 x 16B chunks per thread per stage) ---------
    const int arow = tid >> 2;                 // 0..63
    const int akg  = (tid & 3) << 2;           // 0,4,8,12
    const float* gA = X + (size_t)(m0 + arow) * HID + akg;

    const float* gB[2];
    int bcol[2], bkg[2];
    #pragma unroll
    for (int q = 0; q < 2; ++q) {
        const int c = q * 256 + tid;
        bcol[q] = c >> 2;                      // 0..127
        bkg[q]  = (c & 3) << 2;
        const int n  = n0 + bcol[q];
        const int wr = (n >> 7) * 192 + (n & 127);   // q|k -> weight row (skip V)
        gB[q] = W + (size_t)wr * HID + bkg[q];
    }

    v8f acc[2][2] = {};

    // prologue: stage 0
    copy16_to_lds(gA, &As[0][arow][akg]);
    copy16_to_lds(gB[0], &Bs[0][bcol[0]][bkg[0]]);
    copy16_to_lds(gB[1], &Bs[0][bcol[1]][bkg[1]]);

    const int NSTAGE = HID / 16;   // 48
    for (int st = 0; st < NSTAGE; ++st) {
        const int cur = st & 1;
        if (st + 1 < NSTAGE) {
            const int ks = (st + 1) * 16;
            copy16_to_lds(gA + ks, &As[1 - cur][arow][akg]);
            copy16_to_lds(gB[0] + ks, &Bs[1 - cur][bcol[0]][bkg[0]]);
            copy16_to_lds(gB[1] + ks, &Bs[1 - cur][bcol[1]][bkg[1]]);
            wait_async_le<3>();    // in-order: current stage's 3 ops landed
        } else {
            wait_async_le<0>();
        }
        __syncthreads();

        #pragma unroll
        for (int k4 = 0; k4 < 16; k4 += 4) {
            v2f a[2], b[2];
            #pragma unroll
            for (int i = 0; i < 2; ++i) {
                a[i].x = As[cur][wm * 32 + i * 16 + lrow][k4 + khalf];
                a[i].y = As[cur][wm * 32 + i * 16 + lrow][k4 + khalf + 1];
            }
            #pragma unroll
            for (int jj = 0; jj < 2; ++jj) {
                b[jj].x = Bs[cur][wn * 32 + jj * 16 + lrow][k4 + khalf];
                b[jj].y = Bs[cur][wn * 32 + jj * 16 + lrow][k4 + khalf + 1];
            }
            #pragma unroll
            for (int i = 0; i < 2; ++i)
                #pragma unroll
                for (int jj = 0; jj < 2; ++jj)
                    acc[i][jj] = __builtin_amdgcn_wmma_f32_16x16x4_f32(
                        false, a[i], false, b[jj], (short)0, acc[i][jj],
                        false, false);
        }
        __syncthreads();
    }

    // --- Epilogue: RoPE (adjacent lanes hold the interleaved pair) --------
    const int mhalf = (lane >> 4) * 8;
    #pragma unroll
    for (int jj = 0; jj < 2; ++jj) {
        const int n   = n0 + wn * 32 + jj * 16 + lrow;
        const int j2  = n & 127;
        const int h   = n >> 7;
        const int j   = j2 & 63;
        const int p   = j >> 1;
        const int par = n & 1;
        float* dst = (j2 >= 64) ? Kw : Qw;
        #pragma unroll
        for (int i = 0; i < 2; ++i) {
            #pragma unroll
            for (int r = 0; r < 8; ++r) {
                const float x  = acc[i][jj][r];
                const float px = __shfl_xor(x, 1, 32);
                const int m  = m0 + wm * 32 + i * 16 + r + mhalf;
                const int s  = m >> 1;      // row = s*B + b, B=2
                const int bb = m & 1;
                const float c  = Cos[s * 32 + p];
                const float sn = Sin[s * 32 + p];
                // even: x0*c - x1*sn ; odd: x0*sn + x1*c
                const float val = par ? fmaf(px, sn, x * c)
                                      : fmaf(x, c, -(px * sn));
                dst[(size_t)((bb * NHEAD + h) * SEQ + s) * HDIM + j] = val;
            }
        }
    }
}

// ===========================================================================
// Kernel 2: scores[bh, s, t] = sum_d Q[bh,s,d] * K[bh,t,d]
// 128x128 score tile per block; whole K-dim (64) async-staged once into LDS
// (2 x 128x68 floats = 70 KB of the 320 KB WGP LDS). 8 waves as 4(M) x 2(N),
// 8 accumulators each, 16 fp32-WMMA k-steps. Output streamed to HBM.
// ===========================================================================
__global__ __launch_bounds__(256) void scores_kernel(
    const float* __restrict__ Qw,
    const float* __restrict__ Kw,
    float* __restrict__ Out)
{
    __shared__ float Qs[128][68];   // [row][k], 16B-aligned rows, 4i-mod-64 banks
    __shared__ float Ks[128][68];   // [col][k]

    const int tid  = threadIdx.x;
    const int lane = tid & 31;
    const int wave = tid >> 5;

    const int bh     = blockIdx.z;
    const int rowblk = blockIdx.y * 128;
    const int colblk = blockIdx.x * 128;

    const float* Qb = Qw + (size_t)bh * SEQ * HDIM;
    const float* Kb = Kw + (size_t)bh * SEQ * HDIM;

    #pragma unroll
    for (int pass = 0; pass < 8; ++pass) {
        const int idx = pass * 256 + tid;
        const int r   = idx >> 4;          // 0..127
        const int fg  = (idx & 15) << 2;   // 0..60 step 4
        copy16_to_lds(Qb + (size_t)(rowblk + r) * HDIM + fg, &Qs[r][fg]);
        copy16_to_lds(Kb + (size_t)(colblk + r) * HDIM + fg, &Ks[r][fg]);
    }
    wait_async_le<0>();
    __syncthreads();

    const int wm = wave >> 1;   // 0..3 : 32 rows
    const int wn = wave & 1;    // 0..1 : 64 cols
    const int lrow  = lane & 15;
    const int khalf = (lane >> 4) << 1;

    v8f acc[2][4] = {};

    #pragma unroll
    for (int k = 0; k < HDIM; k += 4) {
        v2f a[2], b[4];
        #pragma unroll
        for (int i = 0; i < 2; ++i) {
            a[i].x = Qs[wm * 32 + i * 16 + lrow][k + khalf];
            a[i].y = Qs[wm * 32 + i * 16 + lrow][k + khalf + 1];
        }
        #pragma unroll
        for (int jj = 0; jj < 4; ++jj) {
            b[jj].x = Ks[wn * 64 + jj * 16 + lrow][k + khalf];
            b[jj].y = Ks[wn * 64 + jj * 16 + lrow][k + khalf + 1];
        }
        #pragma unroll
        for (int i = 0; i < 2; ++i)
            #pragma unroll
            for (int jj = 0; jj < 4; ++jj)
                acc[i][jj] = __builtin_amdgcn_wmma_f32_16x16x4_f32(
                    false, a[i], false, b[jj], (short)0, acc[i][jj],
                    false, false);
    }

    float* Ob = Out + (size_t)bh * SEQ * SEQ;
    const int mhalf = (lane >> 4) * 8;
    #pragma unroll
    for (int i = 0; i < 2; ++i) {
        #pragma unroll
        for (int jj = 0; jj < 4; ++jj) {
            const int col = colblk + wn * 64 + jj * 16 + lrow;
            #pragma unroll
            for (int r = 0; r < 8; ++r) {
                const int m = rowblk + wm * 32 + i * 16 + r + mhalf;
                Ob[(size_t)m * SEQ + col] = acc[i][jj][r];
            }
        }
    }
}

// ===========================================================================
extern "C" void kernel_launch(void* const* d_in, const int* in_sizes, int n_in,
                              void* d_out, int out_size, void* d_ws, size_t ws_size,
                              hipStream_t stream) {
    const float* X   = (const float*)d_in[0];   // (2048, 2, 768)
    const float* W   = (const float*)d_in[1];   // (2304, 768)
    const float* Cos = (const float*)d_in[2];   // (4096, 32)
    const float* Sin = (const float*)d_in[3];   // (4096, 32)
    float* Out = (float*)d_out;                 // (24, 2048, 2048) fp32

    float* Qw = (float*)d_ws;                            // (24, 2048, 64)
    float* Kw = Qw + (size_t)BATCH * NHEAD * SEQ * HDIM; // (24, 2048, 64)

    dim3 gA(M1 / 64, NPACK / 128);      // (64, 12)
    qkv_rope_kernel<<<gA, 256, 0, stream>>>(X, W, Cos, Sin, Qw, Kw);

    dim3 gB(SEQ / 128, SEQ / 128, BATCH * NHEAD);   // (16, 16, 24)
    scores_kernel<<<gB, 256, 0, stream>>>(Qw, Kw, Out);
}